// TokenTransform3D_22342419873946
// MI455X (gfx1250) — compile-verified
//
#include <hip/hip_runtime.h>
#include <hip/hip_bf16.h>

// 1-D vector quantization: out[i] = argmin_j (w[i] - cb[j])^2, out[N] = y.
// VALU-bound (N*K = 268M compare pairs, ~2MB total HBM traffic -> memory is
// ~90ns at 23.3 TB/s, irrelevant). Stage codebook (4KB) in LDS via the CDNA5
// async global->LDS path, then brute-force with exact fp32 reference rounding.

#define CB_K 1024
#define WPT 4   // weights per thread

// The gfx1250 async-LDS builtin takes int4* operands (global src, LDS dst).
typedef int v4i __attribute__((ext_vector_type(4)));
typedef __attribute__((address_space(1))) v4i gv4i;
typedef __attribute__((address_space(3))) v4i lv4i;

// One distance + argmin-update step, bit-exact vs jnp: d = fl(fl(x-c)^2),
// strict '<' with ascending j == first-occurrence argmin tie-breaking.
#define VQ_STEP(X, B, I, C, J)                    \
    {                                             \
        float d_ = (X) - (C);                     \
        d_ *= d_;                                 \
        bool p_ = d_ < (B);                       \
        (I) = p_ ? (J) : (I);                     \
        (B) = p_ ? d_ : (B);                      \
    }

__global__ __launch_bounds__(256) void vq_argmin_kernel(
    const float* __restrict__ weights,
    const float* __restrict__ codebook,
    const int* __restrict__ yscalar,
    float* __restrict__ out,
    int n)
{
    __shared__ __align__(16) float cb[CB_K];
    const int tid = threadIdx.x;

    // Stage the 4KB codebook into LDS. 256 threads x 16B = 4096B in one shot.
#if defined(__gfx1250__) && __has_builtin(__builtin_amdgcn_global_load_async_to_lds_b128)
    {
        gv4i* gsrc = (gv4i*)(codebook + tid * 4);
        lv4i* ldst = (lv4i*)(cb + tid * 4);
        __builtin_amdgcn_global_load_async_to_lds_b128(gsrc, ldst, /*offset=*/0, /*cpol=*/0);
#if __has_builtin(__builtin_amdgcn_s_wait_asynccnt)
        __builtin_amdgcn_s_wait_asynccnt(0);
#else
        asm volatile("s_wait_asynccnt 0" ::: "memory");
#endif
    }
#else
    {
        float4 v = ((const float4*)codebook)[tid];
        ((float4*)cb)[tid] = v;
    }
#endif
    __syncthreads();

    const int base = (int)(blockIdx.x * blockDim.x + tid) * WPT;
    if (base + WPT > n) return;

    float x0 = weights[base + 0];
    float x1 = weights[base + 1];
    float x2 = weights[base + 2];
    float x3 = weights[base + 3];

    float b0 = __builtin_inff(), b1 = b0, b2 = b0, b3 = b0;
    int   i0 = 0, i1 = 0, i2 = 0, i3 = 0;

    // 16 distance/compare pairs per ds_load_b128 (uniform address -> broadcast,
    // no LDS bank conflicts). ~80 VALU ops per 4 codebook entries per thread.
    #pragma unroll 2
    for (int j = 0; j < CB_K; j += 4) {
        float4 c = *(const float4*)&cb[j];

        VQ_STEP(x0, b0, i0, c.x, j + 0)
        VQ_STEP(x1, b1, i1, c.x, j + 0)
        VQ_STEP(x2, b2, i2, c.x, j + 0)
        VQ_STEP(x3, b3, i3, c.x, j + 0)

        VQ_STEP(x0, b0, i0, c.y, j + 1)
        VQ_STEP(x1, b1, i1, c.y, j + 1)
        VQ_STEP(x2, b2, i2, c.y, j + 1)
        VQ_STEP(x3, b3, i3, c.y, j + 1)

        VQ_STEP(x0, b0, i0, c.z, j + 2)
        VQ_STEP(x1, b1, i1, c.z, j + 2)
        VQ_STEP(x2, b2, i2, c.z, j + 2)
        VQ_STEP(x3, b3, i3, c.z, j + 2)

        VQ_STEP(x0, b0, i0, c.w, j + 3)
        VQ_STEP(x1, b1, i1, c.w, j + 3)
        VQ_STEP(x2, b2, i2, c.w, j + 3)
        VQ_STEP(x3, b3, i3, c.w, j + 3)
    }

    out[base + 0] = (float)i0;
    out[base + 1] = (float)i1;
    out[base + 2] = (float)i2;
    out[base + 3] = (float)i3;

    // Second tuple element: passthrough scalar y.
    if (base == 0) {
        out[n] = (float)yscalar[0];
    }
}

extern "C" void kernel_launch(void* const* d_in, const int* in_sizes, int n_in,
                              void* d_out, int out_size, void* d_ws, size_t ws_size,
                              hipStream_t stream) {
    const float* weights  = (const float*)d_in[0];
    const float* codebook = (const float*)d_in[1];
    const int*   yscalar  = (const int*)d_in[2];
    float*       out      = (float*)d_out;

    const int n = in_sizes[0];          // 262144
    const int threads = n / WPT;        // 65536
    const int block = 256;
    const int grid = (threads + block - 1) / block;   // 256 blocks, 8 waves each

    hipLaunchKernelGGL(vq_argmin_kernel, dim3(grid), dim3(block), 0, stream,
                       weights, codebook, yscalar, out, n);
}